// Wkv7_57372173140343
// MI455X (gfx1250) — compile-verified
//
#include <hip/hip_runtime.h>
#include <hip/hip_bf16.h>

// ---------------------------------------------------------------------------
// RWKV-7 (WKV7) chunked-scan implementation for MI455X (gfx1250, wave32).
//
//   M_t = diag(w_t) + a_t b_t^T        (per head, N=64)
//   S_t = S_{t-1} M_t + v_t k_t^T
//   x_t = S_t r_t
//
// Pass A: per chunk (L=64 steps) compute P_c = M_1..M_L and
//         Q_c = sum_s v_s k_s^T M_{s+1}..M_L   (2048-way parallel)
// Pass B: per head, serial over chunks: S_c = S_{c-1} @ P_c + Q_c using
//         v_wmma_f32_16x16x4_f32; P/Q staged to LDS with the Tensor Data
//         Mover, double-buffered so the DMA of chunk c+1 overlaps the
//         WMMAs of chunk c. Saves chunk-start states for pass C.
// Pass C: per chunk, replay L steps from saved start state, emit x_t.
// ---------------------------------------------------------------------------

#define Tt 4096
#define Hh 32
#define Nn 64
#define Dd (Hh * Nn)   // 2048
#define Ll 64          // chunk length
#define Cc (Tt / Ll)   // 64 chunks

typedef __attribute__((ext_vector_type(2))) float v2f;
typedef __attribute__((ext_vector_type(8))) float v8f;
typedef __attribute__((ext_vector_type(4))) unsigned int u32x4;
typedef __attribute__((ext_vector_type(4))) int i32x4;
typedef __attribute__((ext_vector_type(8))) int i32x8;

#if defined(__has_builtin)
#if __has_builtin(__builtin_amdgcn_tensor_load_to_lds) && \
    __has_builtin(__builtin_amdgcn_s_wait_tensorcnt)
#define USE_TDM 1
#endif
#endif
#ifndef USE_TDM
#define USE_TDM 0
#endif

#if USE_TDM
// Async-copy nElems contiguous f32 from global memory to LDS via the TDM.
// D# per CDNA5 ISA: 2D tensor, one row of nElems
// 4-byte elements; groups 2/3 unused (tile_dim2 = 0).
__device__ __forceinline__ void tdm_load_1d(unsigned lds_off,
                                            const float* gptr, int nElems) {
  const unsigned long long ga = (unsigned long long)(size_t)gptr;
  u32x4 g0;
  g0[0] = 1u;                                   // count=1, user descriptor
  g0[1] = lds_off;                              // lds_addr (bytes)
  g0[2] = (unsigned)(ga & 0xFFFFFFFFu);         // global_addr[31:0]
  g0[3] = (unsigned)((ga >> 32) & 0x01FFFFFFu)  // global_addr[56:32]
          | (2u << 30);                         // type = 2 (tensor/image)
  i32x8 g1;
  g1[0] = (2 << 16);                            // data_size=2 -> 4 bytes
  g1[1] = (nElems & 0xFFFF) << 16;              // tensor_dim0[15:0]  (b63:48)
  g1[2] = ((nElems >> 16) & 0xFFFF)             // tensor_dim0[31:16] (b79:64)
          | (1 << 16);                          // tensor_dim1 = 1    (b95:80)
  g1[3] = (nElems & 0xFFFF) << 16;              // tile_dim0          (b127:112)
  g1[4] = 1;                                    // tile_dim1=1, tile_dim2=0
  g1[5] = nElems;                               // tensor_dim0_stride[31:0]
  g1[6] = 0;                                    // stride0[47:32], stride1[15:0]
  g1[7] = 0;
  i32x4 gz = {0, 0, 0, 0};
#if __clang_major__ >= 23
  i32x8 gz8 = {0, 0, 0, 0, 0, 0, 0, 0};
  __builtin_amdgcn_tensor_load_to_lds(g0, g1, gz, gz, gz8, 0);
#else
  __builtin_amdgcn_tensor_load_to_lds(g0, g1, gz, gz, 0);
#endif
}
#endif

// ---------------------------------------------------------------------------
// Pass A: build per-chunk transition product P and additive term Q.
// Block = 64 threads (2 waves); thread i owns row i of P and Q in VGPRs.
// P <- P*diag(w) + (P a) b^T ; Q <- Q*diag(w) + (Q a) b^T + v k^T
// ---------------------------------------------------------------------------
__global__ __launch_bounds__(64) void wkv7_pq(
    const float* __restrict__ w, const float* __restrict__ k,
    const float* __restrict__ v, const float* __restrict__ a,
    const float* __restrict__ b, float* __restrict__ Pg,
    float* __restrict__ Qg) {
  const int c = blockIdx.x, h = blockIdx.y, i = threadIdx.x;
  __shared__ float sw[Nn], sk[Nn], sv[Nn], sa[Nn], sb[Nn];

  float Prow[Nn], Qrow[Nn];
#pragma unroll
  for (int j = 0; j < Nn; ++j) {
    Prow[j] = (j == i) ? 1.0f : 0.0f;
    Qrow[j] = 0.0f;
  }

  const int t0 = c * Ll;
  for (int s = 0; s < Ll; ++s) {
    const int base = (t0 + s) * Dd + h * Nn + i;
    sw[i] = w[base]; sk[i] = k[base]; sv[i] = v[base];
    sa[i] = a[base]; sb[i] = b[base];
    if (s + 1 < Ll) {  // stream next timestep into near caches
      __builtin_prefetch(w + base + Dd, 0, 0);
      __builtin_prefetch(k + base + Dd, 0, 0);
      __builtin_prefetch(a + base + Dd, 0, 0);
      __builtin_prefetch(b + base + Dd, 0, 0);
    }
    __syncthreads();

    float pa = 0.0f, qa = 0.0f;
#pragma unroll
    for (int j = 0; j < Nn; ++j) {
      pa += Prow[j] * sa[j];
      qa += Qrow[j] * sa[j];
    }
    const float vi = sv[i];
#pragma unroll
    for (int j = 0; j < Nn; ++j) {
      Prow[j] = Prow[j] * sw[j] + pa * sb[j];
      Qrow[j] = Qrow[j] * sw[j] + qa * sb[j] + vi * sk[j];
    }
    __syncthreads();
  }

  const size_t off = ((size_t)(h * Cc + c)) * Nn * Nn + (size_t)i * Nn;
#pragma unroll
  for (int j = 0; j < Nn; ++j) {
    Pg[off + j] = Prow[j];
    Qg[off + j] = Qrow[j];
  }
}

// ---------------------------------------------------------------------------
// Pass B: serial chunk chain per head: S = S @ P_c + Q_c via WMMA f32.
// Block = 256 threads (8 waves). S + double-buffered P/Q in LDS (80 KB).
// TDM prefetches chunk c+1 while chunk c computes (wave 0 issues/waits).
// ---------------------------------------------------------------------------
__global__ __launch_bounds__(256) void wkv7_chain(
    const float* __restrict__ state2, const float* __restrict__ Pg,
    const float* __restrict__ Qg, float* __restrict__ Sst,
    float* __restrict__ state_out) {
  const int h = blockIdx.x;
  const int tid = threadIdx.x;
  const int lane = tid & 31;
  const int wave = tid >> 5;

  __shared__ float S[Nn * Nn];
  __shared__ float Pt[2][Nn * Nn];
  __shared__ float Qt[2][Nn * Nn];

  for (int e = tid; e < Nn * Nn; e += 256) S[e] = state2[(size_t)h * Nn * Nn + e];

#if USE_TDM
  if (wave == 0) {  // wave-uniform branch; TDM ignores EXEC
    const size_t off0 = ((size_t)h * Cc) * Nn * Nn;
    tdm_load_1d((unsigned)(size_t)&Pt[0][0], Pg + off0, Nn * Nn);
    tdm_load_1d((unsigned)(size_t)&Qt[0][0], Qg + off0, Nn * Nn);
    __builtin_amdgcn_s_wait_tensorcnt(0);
  }
#else
  {
    const size_t off0 = ((size_t)h * Cc) * Nn * Nn;
    for (int e = tid; e < Nn * Nn; e += 256) {
      Pt[0][e] = Pg[off0 + e];
      Qt[0][e] = Qg[off0 + e];
    }
  }
#endif
  __syncthreads();

  // Per-lane WMMA coordinates (ISA 16x16x4 f32 layouts):
  // A: row = lane&15, VGPR0/1 = K {kb, kb+1}, kb = 2*(lane>=16)
  // B: col = lane&15, VGPR0/1 = rows {kb, kb+1}
  // C/D: col = lane&15, VGPR v = row v + 8*(lane>=16)
  const int lc = lane & 15;
  const int kb = (lane >> 4) * 2;
  const int mhi = (lane >> 4) * 8;

  for (int c = 0; c < Cc; ++c) {
    const int buf = c & 1;
    const size_t off = ((size_t)(h * Cc + c)) * Nn * Nn;

#if USE_TDM
    if (wave == 0 && c + 1 < Cc) {  // async prefetch of next chunk's P/Q
      tdm_load_1d((unsigned)(size_t)&Pt[buf ^ 1][0], Pg + off + Nn * Nn,
                  Nn * Nn);
      tdm_load_1d((unsigned)(size_t)&Qt[buf ^ 1][0], Qg + off + Nn * Nn,
                  Nn * Nn);
    }
#else
    if (c + 1 < Cc) {
      for (int e = tid; e < Nn * Nn; e += 256) {
        Pt[buf ^ 1][e] = Pg[off + Nn * Nn + e];
        Qt[buf ^ 1][e] = Qg[off + Nn * Nn + e];
      }
    }
#endif

    // Save chunk-start state (needed by pass C).
    for (int e = tid; e < Nn * Nn; e += 256) Sst[off + e] = S[e];

    v8f acc[2];
#pragma unroll
    for (int tt = 0; tt < 2; ++tt) {
      const int tile = wave * 2 + tt;  // 16 tiles total, 2 per wave
      const int mi = (tile >> 2) * 16;
      const int nj = (tile & 3) * 16;

      v8f cacc;
#pragma unroll
      for (int vv = 0; vv < 8; ++vv)
        cacc[vv] = Qt[buf][(mi + mhi + vv) * Nn + nj + lc];

#pragma unroll
      for (int kk = 0; kk < Nn; kk += 4) {
        v2f av, bv;
        av[0] = S[(mi + lc) * Nn + kk + kb];
        av[1] = S[(mi + lc) * Nn + kk + kb + 1];
        bv[0] = Pt[buf][(kk + kb) * Nn + nj + lc];
        bv[1] = Pt[buf][(kk + kb + 1) * Nn + nj + lc];
        cacc = __builtin_amdgcn_wmma_f32_16x16x4_f32(
            /*neg_a=*/false, av, /*neg_b=*/false, bv,
            /*c_mod=*/(short)0, cacc, /*reuse_a=*/false, /*reuse_b=*/false);
      }
      acc[tt] = cacc;
    }

    __syncthreads();  // all waves done reading S before overwrite
#pragma unroll
    for (int tt = 0; tt < 2; ++tt) {
      const int tile = wave * 2 + tt;
      const int mi = (tile >> 2) * 16;
      const int nj = (tile & 3) * 16;
#pragma unroll
      for (int vv = 0; vv < 8; ++vv)
        S[(mi + mhi + vv) * Nn + nj + lc] = acc[tt][vv];
    }

#if USE_TDM
    if (wave == 0 && c + 1 < Cc) __builtin_amdgcn_s_wait_tensorcnt(0);
#endif
    __syncthreads();  // publishes S writeback + next chunk's P/Q to all waves
  }

  for (int e = tid; e < Nn * Nn; e += 256)
    state_out[(size_t)h * Nn * Nn + e] = S[e];
}

// ---------------------------------------------------------------------------
// Pass C: replay each chunk from its saved start state; emit x_t = S_t r_t.
// Block = 64 threads; thread i owns row i of S in VGPRs.
// ---------------------------------------------------------------------------
__global__ __launch_bounds__(64) void wkv7_out(
    const float* __restrict__ r, const float* __restrict__ w,
    const float* __restrict__ k, const float* __restrict__ v,
    const float* __restrict__ a, const float* __restrict__ b,
    const float* __restrict__ Sst, float* __restrict__ x) {
  const int c = blockIdx.x, h = blockIdx.y, i = threadIdx.x;
  __shared__ float sr[Nn], sw[Nn], sk[Nn], sv[Nn], sa[Nn], sb[Nn];

  float S[Nn];
  const size_t soff = ((size_t)(h * Cc + c)) * Nn * Nn + (size_t)i * Nn;
#pragma unroll
  for (int j = 0; j < Nn; ++j) S[j] = Sst[soff + j];

  const int t0 = c * Ll;
  for (int s = 0; s < Ll; ++s) {
    const int base = (t0 + s) * Dd + h * Nn + i;
    sr[i] = r[base]; sw[i] = w[base]; sk[i] = k[base];
    sv[i] = v[base]; sa[i] = a[base]; sb[i] = b[base];
    if (s + 1 < Ll) {
      __builtin_prefetch(r + base + Dd, 0, 0);
      __builtin_prefetch(v + base + Dd, 0, 0);
    }
    __syncthreads();

    float sai = 0.0f;
#pragma unroll
    for (int j = 0; j < Nn; ++j) sai += S[j] * sa[j];
    const float vi = sv[i];
    float xi = 0.0f;
#pragma unroll
    for (int j = 0; j < Nn; ++j) {
      S[j] = S[j] * sw[j] + sai * sb[j] + vi * sk[j];
      xi += S[j] * sr[j];
    }
    x[base] = xi;  // x is (T,H,1,N) -> same flat layout as (T,D)
    __syncthreads();
  }
}

// ---------------------------------------------------------------------------
// Launch: inputs = {seq_length, r, w, k, v, a, b, state2} (all f32 but [0]).
// d_out = x (T*D floats) ++ state_out (H*N*N floats).
// Workspace: P (H*C*N*N) ++ Q (same) ++ chunk-start states (same) = ~96 MB.
// ---------------------------------------------------------------------------
extern "C" void kernel_launch(void* const* d_in, const int* in_sizes, int n_in,
                              void* d_out, int out_size, void* d_ws, size_t ws_size,
                              hipStream_t stream) {
  const float* r  = (const float*)d_in[1];
  const float* w  = (const float*)d_in[2];
  const float* k  = (const float*)d_in[3];
  const float* v  = (const float*)d_in[4];
  const float* a  = (const float*)d_in[5];
  const float* b  = (const float*)d_in[6];
  const float* s2 = (const float*)d_in[7];

  float* x         = (float*)d_out;
  float* state_out = x + (size_t)Tt * Dd;

  const size_t blk = (size_t)Hh * Cc * Nn * Nn;
  float* Pg  = (float*)d_ws;
  float* Qg  = Pg + blk;
  float* Sst = Qg + blk;

  dim3 gridA(Cc, Hh);
  wkv7_pq<<<gridA, 64, 0, stream>>>(w, k, v, a, b, Pg, Qg);
  wkv7_chain<<<Hh, 256, 0, stream>>>(s2, Pg, Qg, Sst, state_out);
  wkv7_out<<<gridA, 64, 0, stream>>>(r, w, k, v, a, b, Sst, x);
}